// BilinearInteraction_47622597378059
// MI455X (gfx1250) — compile-verified
//
#include <hip/hip_runtime.h>

typedef __attribute__((ext_vector_type(2))) float v2f;
typedef __attribute__((ext_vector_type(4))) float v4f;
typedef __attribute__((ext_vector_type(8))) float v8f;

#define NUM_FIELDS 32
#define D_EMBED    64
#define BATCH_SZ   2048
#define NUM_PAIRS  496
#define BM         64           // batch rows per workgroup (4 waves x 16)
#define SW_STRIDE  160          // floats per kp-row: 160 mod 64 == 32 -> lane halves hit disjoint banks
#define SA_STRIDE  68           // A tile stride: b64 fragment reads cover all 64 banks once

__global__ __launch_bounds__(128)
void bilinear_wmma_kernel(const float* __restrict__ femb,
                          const float* __restrict__ W,
                          float* __restrict__ out)
{
    // sW2[kp*160 + 2e + z] = W[p][e][2kp+z]  (transposed, K-pair interleaved)
    __shared__ float sW2[(D_EMBED / 2) * SW_STRIDE];   // 20 KB
    __shared__ float sA[BM * SA_STRIDE];               // 17 KB

    const int p  = blockIdx.y;
    const int b0 = blockIdx.x * BM;
    const int t  = threadIdx.x;

    // decode pair p -> (fi, fj) in itertools.combinations order (uniform / scalar)
    int rem = p, fi = 0;
    while (rem >= NUM_FIELDS - 1 - fi) { rem -= NUM_FIELDS - 1 - fi; ++fi; }
    const int fj = fi + 1 + rem;

    // ---- Stage W[p] transposed+interleaved into LDS ----
    // lanes -> consecutive e: LDS b64 writes cover all 64 banks once (conflict-free);
    // global reads are 8B/row scattered but W[p] (16KB) is L2-resident.
    {
        const float* Wp = W + (size_t)p * (D_EMBED * D_EMBED);
        #pragma unroll
        for (int it = 0; it < 16; ++it) {
            int q  = it * 128 + t;      // 0..2047 float2 index
            int e  = q & 63;
            int d2 = q >> 6;            // kp = 0..31
            v2f w = *(const v2f*)(Wp + e * D_EMBED + 2 * d2);
            *(v2f*)(sW2 + d2 * SW_STRIDE + 2 * e) = w;
        }
    }
    // ---- Stage A tile: 64 rows of field fi (fully coalesced float4) ----
    {
        #pragma unroll
        for (int it = 0; it < 8; ++it) {
            int q = it * 128 + t;       // 0..1023
            int m = q >> 4;
            int c = (q & 15) << 2;
            v4f v = *(const v4f*)(femb + (size_t)(b0 + m) * (NUM_FIELDS * D_EMBED)
                                        + fi * D_EMBED + c);
            *(v4f*)(sA + m * SA_STRIDE + c) = v;    // 68*4=272 B, multiple of 16
        }
    }
    __syncthreads();

    const int wave = t >> 5;
    const int lane = t & 31;
    const int lh   = lane >> 4;     // lane half: K offset +0 / +2 (kp offset +0 / +1)
    const int l15  = lane & 15;

    v8f acc0 = {}, acc1 = {}, acc2 = {}, acc3 = {};

    const float* aPtr = sA  + (wave * 16 + l15) * SA_STRIDE + 2 * lh;  // A[M=l15][K=4s+2lh..]
    const float* bPtr = sW2 + lh * SW_STRIDE + 2 * l15;                // B[kp=2s+lh][N=l15]

    #pragma unroll
    for (int s = 0; s < 16; ++s) {
        // A fragment 16x4: v2f = {A[M][4s+2lh], A[M][4s+2lh+1]}
        v2f a = *(const v2f*)(aPtr + 4 * s);
        const float* bk = bPtr + s * (2 * SW_STRIDE);
        // B fragments: one contiguous ds_load_b64 each -> lands in adjacent VGPR pair
        v2f bv0 = *(const v2f*)(bk +  0);
        v2f bv1 = *(const v2f*)(bk + 32);
        v2f bv2 = *(const v2f*)(bk + 64);
        v2f bv3 = *(const v2f*)(bk + 96);
        acc0 = __builtin_amdgcn_wmma_f32_16x16x4_f32(false, a, false, bv0, (short)0, acc0, false, false);
        acc1 = __builtin_amdgcn_wmma_f32_16x16x4_f32(false, a, false, bv1, (short)0, acc1, false, false);
        acc2 = __builtin_amdgcn_wmma_f32_16x16x4_f32(false, a, false, bv2, (short)0, acc2, false, false);
        acc3 = __builtin_amdgcn_wmma_f32_16x16x4_f32(false, a, false, bv3, (short)0, acc3, false, false);
    }

    // ---- Epilogue: D[r] holds M = r + 8*lh, N = nt*16 + l15 ----
    const size_t outRow = (size_t)NUM_PAIRS * D_EMBED;
    #pragma unroll
    for (int r = 0; r < 8; ++r) {
        size_t b = (size_t)b0 + wave * 16 + r + 8 * lh;
        const float* vjRow = femb + b * (NUM_FIELDS * D_EMBED) + fj * D_EMBED + l15;
        float* oRow = out + b * outRow + (size_t)p * D_EMBED + l15;
        float vj0 = vjRow[0], vj1 = vjRow[16], vj2 = vjRow[32], vj3 = vjRow[48];
        __builtin_nontemporal_store(acc0[r] * vj0, oRow +  0);
        __builtin_nontemporal_store(acc1[r] * vj1, oRow + 16);
        __builtin_nontemporal_store(acc2[r] * vj2, oRow + 32);
        __builtin_nontemporal_store(acc3[r] * vj3, oRow + 48);
    }
}

extern "C" void kernel_launch(void* const* d_in, const int* in_sizes, int n_in,
                              void* d_out, int out_size, void* d_ws, size_t ws_size,
                              hipStream_t stream) {
    const float* femb = (const float*)d_in[0];   // (2048, 32, 64) f32
    const float* W    = (const float*)d_in[1];   // (496, 64, 64) f32
    float* out        = (float*)d_out;           // (2048, 496, 64) f32

    dim3 grid(BATCH_SZ / BM, NUM_PAIRS);
    bilinear_wmma_kernel<<<grid, 128, 0, stream>>>(femb, W, out);
}